// BandSplitModule_19018115187349
// MI455X (gfx1250) — compile-verified
//
#include <hip/hip_runtime.h>

typedef __attribute__((ext_vector_type(2))) float v2f;
typedef __attribute__((ext_vector_type(8))) float v8f;

#define KBANDS 36
#define EMBED  128
#define MAXF   130
#define TN     64      // t-columns per block
#define TDIM   2048
#define NBTOT  1025
#define FPAD   132     // max f rounded up to multiple of 4
#define PAIRS  (FPAD/2)
#define PSTR   (2*TN)  // floats per K-pair row group

// LDS layout: element (row r, col t) lives at xs[(r>>1)*PSTR + 2*t + (r&1)]
// -> the (K, K+1) pair needed by a WMMA B fragment is one aligned b64.

__global__ __launch_bounds__(256) void bandsplit_ln_gemm_kernel(
    const float* __restrict__ spec,   // (B, NB, T, 2)
    const float* __restrict__ gamma,  // (K, MAXF)
    const float* __restrict__ beta,   // (K, MAXF)
    const float* __restrict__ W,      // (K, EMBED, MAXF)
    const float* __restrict__ bias,   // (K, EMBED)
    float* __restrict__ out)          // (B, EMBED, K, T)
{
    __shared__ float xs[PAIRS * PSTR];
    __shared__ float ps [256];
    __shared__ float pss[256];
    __shared__ float mu_s[TN];
    __shared__ float rs_s[TN];

    const int tid  = (int)threadIdx.x;
    const int k    = (int)blockIdx.y;
    const int col0 = (int)blockIdx.x * TN;
    const int b    = col0 >> 11;          // T = 2048
    const int t0   = col0 & (TDIM - 1);

    int start, nb;
    if      (k < 20) { start = k * 16;            nb = 16; }
    else if (k < 30) { start = 320 + (k-20)*32;   nb = 32; }
    else if (k < 35) { start = 640 + (k-30)*64;   nb = 64; }
    else             { start = 960;               nb = 65; }
    const int f   = 2 * nb;
    const int fp2 = nb;                   // filled K-pairs

    // ---- Fill x tile with async global->LDS b64 copies (ASYNCcnt path) ----
    // (c0,c1) at (b, start+bin, t0+tl) is 8B aligned in global and lands at the
    // contiguous pair slot xs[bin*PSTR + 2*tl] in LDS.
    for (int idx = tid; idx < nb * TN; idx += 256) {
        int bin = idx >> 6;               // pair index == bin
        int tl  = idx & (TN - 1);
        const float* gp = spec + ((size_t)(b * NBTOT + start + bin) * TDIM + (t0 + tl)) * 2;
        unsigned lds_off = (unsigned)(uintptr_t)(xs + bin * PSTR + 2 * tl); // addr[31:0] = LDS offset
        asm volatile("global_load_async_to_lds_b64 %0, %1, off"
                     :: "v"(lds_off), "v"(gp) : "memory");
    }
    // zero-pad pairs fp2..PAIRS so the ragged K tail multiplies against zeros
    for (int idx = fp2 * TN + tid; idx < PAIRS * TN; idx += 256) {
        v2f z = {0.0f, 0.0f};
        *(v2f*)(xs + idx * 2) = z;        // idx = pair*TN + tl -> offset pair*PSTR + 2*tl
    }
    asm volatile("s_wait_asynccnt 0" ::: "memory");
    __syncthreads();

    // ---- Per-column LayerNorm statistics (4-way parallel over pairs) ----
    {
        const int col  = tid & (TN - 1);
        const int part = tid >> 6;        // 0..3
        float s = 0.0f, ss = 0.0f;
        for (int p = part; p < fp2; p += 4) {
            v2f v = *(const v2f*)(xs + p * PSTR + 2 * col);  // b64, conflict-free
            s  += v.x + v.y;
            ss += v.x * v.x + v.y * v.y;
        }
        ps [tid] = s;
        pss[tid] = ss;
    }
    __syncthreads();
    if (tid < TN) {
        float s  = ps [tid] + ps [tid + 64] + ps [tid + 128] + ps [tid + 192];
        float ss = pss[tid] + pss[tid + 64] + pss[tid + 128] + pss[tid + 192];
        float inv = 1.0f / (float)f;
        float mu  = s * inv;
        float var = ss * inv - mu * mu;
        mu_s[tid] = mu;
        rs_s[tid] = rsqrtf(var + 1e-5f);
    }
    __syncthreads();

    // ---- Normalize in place (pairs < fp2 only; pad pairs stay zero) ----
    for (int idx = tid; idx < fp2 * TN; idx += 256) {
        int p  = idx >> 6;
        int tl = idx & (TN - 1);
        v2f g  = *(const v2f*)(gamma + k * MAXF + 2 * p);   // aligned: k*130 + even
        v2f bt = *(const v2f*)(beta  + k * MAXF + 2 * p);
        float mu = mu_s[tl], rs = rs_s[tl];
        v2f v = *(v2f*)(xs + p * PSTR + 2 * tl);
        v.x = (v.x - mu) * rs * g.x + bt.x;
        v.y = (v.y - mu) * rs * g.y + bt.y;
        *(v2f*)(xs + p * PSTR + 2 * tl) = v;
    }
    __syncthreads();

    // ---- GEMM: D[e,t] = sum_f W[k,e,f] * xn[f,t]  via V_WMMA_F32_16X16X4_F32 ----
    const int wave = tid >> 5;
    const int lane = tid & 31;
    const int half = lane >> 4;   // 0: K+0/1 pair, 1: K+2/3 pair (32-bit A/B layout)
    const int l16  = lane & 15;
    const int e0   = wave * 16;

    v8f acc0 = {}, acc1 = {}, acc2 = {}, acc3 = {};
    const int ksteps = (f + 3) >> 2;
    const float* wrow = W + ((size_t)k * EMBED + (e0 + l16)) * MAXF;  // row M = l16

    // Main loop: no clamp needed (fb+1 < f guaranteed for s4 < ksteps-1)
    // -> W access is pure base + immediate offset.
    #pragma unroll 2
    for (int s4 = 0; s4 < ksteps - 1; ++s4) {
        v2f a = *(const v2f*)(wrow + s4 * 4 + half * 2);

        const float* xr = xs + (s4 * 2 + half) * PSTR + 2 * l16;
        v2f b0 = *(const v2f*)(xr      );
        v2f b1 = *(const v2f*)(xr + 32 );   // +16 columns
        v2f b2 = *(const v2f*)(xr + 64 );   // +32 columns
        v2f b3 = *(const v2f*)(xr + 96 );   // +48 columns

        acc0 = __builtin_amdgcn_wmma_f32_16x16x4_f32(false, a, false, b0, (short)0, acc0, false, false);
        acc1 = __builtin_amdgcn_wmma_f32_16x16x4_f32(false, a, false, b1, (short)0, acc1, false, false);
        acc2 = __builtin_amdgcn_wmma_f32_16x16x4_f32(false, a, false, b2, (short)0, acc2, false, false);
        acc3 = __builtin_amdgcn_wmma_f32_16x16x4_f32(false, a, false, b3, (short)0, acc3, false, false);
    }

    // Peeled final step: clamp only here (f=130 band tail). Matching B rows are
    // zero in LDS, so any finite A value is harmless -> branch/EXEC-free select.
    {
        int s4 = ksteps - 1;
        int fb = s4 * 4 + half * 2;
        int fbc = (fb + 1 < f) ? fb : 0;
        v2f a = *(const v2f*)(wrow + fbc);

        const float* xr = xs + (s4 * 2 + half) * PSTR + 2 * l16;
        v2f b0 = *(const v2f*)(xr      );
        v2f b1 = *(const v2f*)(xr + 32 );
        v2f b2 = *(const v2f*)(xr + 64 );
        v2f b3 = *(const v2f*)(xr + 96 );

        acc0 = __builtin_amdgcn_wmma_f32_16x16x4_f32(false, a, false, b0, (short)0, acc0, false, false);
        acc1 = __builtin_amdgcn_wmma_f32_16x16x4_f32(false, a, false, b1, (short)0, acc1, false, false);
        acc2 = __builtin_amdgcn_wmma_f32_16x16x4_f32(false, a, false, b2, (short)0, acc2, false, false);
        acc3 = __builtin_amdgcn_wmma_f32_16x16x4_f32(false, a, false, b3, (short)0, acc3, false, false);
    }

    // ---- Store D + bias: C/D layout -> VGPR r holds (M = r + 8*half, N = l16) ----
    #pragma unroll
    for (int r = 0; r < 8; ++r) {
        int e = e0 + r + half * 8;
        float bv = bias[k * EMBED + e];
        size_t base = ((size_t)(b * EMBED + e) * KBANDS + k) * TDIM + (t0 + l16);
        out[base     ] = acc0[r] + bv;
        out[base + 16] = acc1[r] + bv;
        out[base + 32] = acc2[r] + bv;
        out[base + 48] = acc3[r] + bv;
    }
}

extern "C" void kernel_launch(void* const* d_in, const int* in_sizes, int n_in,
                              void* d_out, int out_size, void* d_ws, size_t ws_size,
                              hipStream_t stream) {
    const float* spec  = (const float*)d_in[0];
    const float* gamma = (const float*)d_in[1];
    const float* beta  = (const float*)d_in[2];
    const float* W     = (const float*)d_in[3];
    const float* bias  = (const float*)d_in[4];
    float* out = (float*)d_out;

    dim3 grid((8 * TDIM) / TN, KBANDS);  // (256, 36)
    bandsplit_ln_gemm_kernel<<<grid, 256, 0, stream>>>(spec, gamma, beta, W, bias, out);
}